// MDCA_42752104465038
// MI455X (gfx1250) — compile-verified
//
#include <hip/hip_runtime.h>
#include <hip/hip_bf16.h>

typedef __attribute__((ext_vector_type(2))) float v2f;
typedef __attribute__((ext_vector_type(8))) float v8f;

#define HH 256
#define WW 256
#define CC 256
#define BB 8
#define MIP 16
#define LTOT 1024  // H + W + H + H

// ---------------------------------------------------------------------------
// Kernel 1: per-(b,c) row means and column means of a 256x256 f32 slab.
// One block per (b,c). 8 waves; wave w owns rows [32w, 32w+32).
// Lanes load two coalesced float4s per row (cols 4l..4l+3 and 128+4l..).
// Column partials live in registers (lane owns the same 8 cols every row);
// row sums use a wave shfl_xor reduction; cross-wave combine via LDS.
// ---------------------------------------------------------------------------
__global__ void reduce_rc(const float* __restrict__ x,
                          float* __restrict__ rowmean,
                          float* __restrict__ colmean) {
  __shared__ float s_row[256];
  __shared__ float s_colp[8 * 256];

  const int bc   = blockIdx.x;               // 0..2047
  const float* xp = x + (size_t)bc * (HH * WW);
  const int tid  = threadIdx.x;
  const int wave = tid >> 5;
  const int lane = tid & 31;

  float colAcc[8];
#pragma unroll
  for (int j = 0; j < 8; ++j) colAcc[j] = 0.0f;

#pragma unroll 4
  for (int i = 0; i < 32; ++i) {
    const int h = wave * 32 + i;
    const float4 v1 = *reinterpret_cast<const float4*>(xp + h * WW + 4 * lane);
    const float4 v2 = *reinterpret_cast<const float4*>(xp + h * WW + 128 + 4 * lane);
    colAcc[0] += v1.x; colAcc[1] += v1.y; colAcc[2] += v1.z; colAcc[3] += v1.w;
    colAcc[4] += v2.x; colAcc[5] += v2.y; colAcc[6] += v2.z; colAcc[7] += v2.w;
    float r = ((v1.x + v1.y) + (v1.z + v1.w)) + ((v2.x + v2.y) + (v2.z + v2.w));
#pragma unroll
    for (int off = 16; off >= 1; off >>= 1)
      r += __shfl_xor(r, off, 32);
    if (lane == 0) s_row[h] = r;
  }

#pragma unroll
  for (int j = 0; j < 4; ++j) {
    s_colp[wave * 256 + 4 * lane + j]       = colAcc[j];
    s_colp[wave * 256 + 128 + 4 * lane + j] = colAcc[4 + j];
  }
  __syncthreads();

  float cs = 0.0f;
#pragma unroll
  for (int w = 0; w < 8; ++w) cs += s_colp[w * 256 + tid];
  colmean[bc * 256 + tid] = cs * (1.0f / 256.0f);
  rowmean[bc * 256 + tid] = s_row[tid] * (1.0f / 256.0f);
}

// ---------------------------------------------------------------------------
// Kernel 2: y[b,m,l] = relu(BN(sum_c cat[b,c,l] * w1[m,c] + b1[m]))
// cat[b,c,l] = rowmean for segs 0,2,3 and colmean for seg 1 (d1==d2==row mean).
// One wave per (b, 16-wide l tile); 64 steps of V_WMMA_F32_16X16X4_F32.
//   A (16x4 f32): lanes 0-15 M=lane, K={k0,k0+1}; lanes 16-31 K={k0+2,k0+3}.
//   B (4x16 f32): lanes 0-15 N=lane, K={k0,k0+1}; lanes 16-31 K={k0+2,k0+3}.
//   C/D: VGPR r -> M = r + 8*(lane/16), N = lane%16.
// ---------------------------------------------------------------------------
__global__ void mlp_wmma(const float* __restrict__ rowmean,
                         const float* __restrict__ colmean,
                         const float* __restrict__ w1,
                         const float* __restrict__ b1,
                         const float* __restrict__ bn_gamma,
                         const float* __restrict__ bn_beta,
                         const float* __restrict__ bn_mean,
                         const float* __restrict__ bn_var,
                         float* __restrict__ y) {
  __shared__ float s_scale[MIP];
  __shared__ float s_shift[MIP];
  const int tid = threadIdx.x;
  if (tid < MIP) {
    const float sc = bn_gamma[tid] * rsqrtf(bn_var[tid] + 1e-5f);
    s_scale[tid] = sc;
    s_shift[tid] = (b1[tid] - bn_mean[tid]) * sc + bn_beta[tid];
  }
  __syncthreads();

  const int wave = tid >> 5;
  const int lane = tid & 31;
  const int gw   = blockIdx.x * 8 + wave;   // 0..511
  const int b    = gw >> 6;                 // 0..7
  const int l0   = (gw & 63) * 16;          // 0..1008
  const int seg  = l0 >> 8;                 // 0..3
  const float* segp = (seg == 1) ? colmean : rowmean;

  const int n     = lane & 15;
  const int khalf = (lane >> 4) << 1;       // 0 or 2

  const float* aptr = w1 + n * CC + khalf;                 // A: m = lane%16
  const float* bptr = segp + (b * CC) * 256 + (l0 & 255) + n;

  v8f acc = {};
#pragma unroll 8
  for (int k0 = 0; k0 < CC; k0 += 4) {
    v2f a;
    a.x = aptr[k0];
    a.y = aptr[k0 + 1];
    v2f bmat;
    bmat.x = bptr[(k0 + khalf) * 256];
    bmat.y = bptr[(k0 + khalf + 1) * 256];
    acc = __builtin_amdgcn_wmma_f32_16x16x4_f32(
        /*neg_a=*/false, a, /*neg_b=*/false, bmat,
        /*c_mod=*/(short)0, acc, /*reuse_a=*/false, /*reuse_b=*/false);
  }

  const int mbase = (lane >> 4) * 8;
#pragma unroll
  for (int r = 0; r < 8; ++r) {
    const int m = mbase + r;
    float v = acc[r] * s_scale[m] + s_shift[m];
    v = v > 0.0f ? v : 0.0f;
    y[(b * MIP + m) * LTOT + l0 + n] = v;
  }
}

// ---------------------------------------------------------------------------
// Kernel 3: per-(b,l) sigmoid gates over the 4 segments.
// ---------------------------------------------------------------------------
__global__ void gates_k(const float* __restrict__ y,
                        const float* __restrict__ wh, const float* __restrict__ bh,
                        const float* __restrict__ ww, const float* __restrict__ bw,
                        const float* __restrict__ wd1, const float* __restrict__ bd1,
                        const float* __restrict__ wd2, const float* __restrict__ bd2,
                        float* __restrict__ g) {
  const int gid = blockIdx.x * blockDim.x + threadIdx.x;   // 0..8191
  const int b   = gid >> 10;
  const int l   = gid & 1023;
  const int seg = l >> 8;
  const float* wsel = (seg == 0) ? wh : (seg == 1) ? ww : (seg == 2) ? wd1 : wd2;
  float s = (seg == 0) ? bh[0] : (seg == 1) ? bw[0] : (seg == 2) ? bd1[0] : bd2[0];
#pragma unroll
  for (int m = 0; m < MIP; ++m)
    s += y[(b * MIP + m) * LTOT + l] * wsel[m];
  g[gid] = 1.0f / (1.0f + expf(-s));
}

// ---------------------------------------------------------------------------
// Kernel 4: imp[b,0,h,w] = a_h[b,h] * a_w[b,w] * a_d1[b,w] * a_d2[b,w]
// ---------------------------------------------------------------------------
__global__ void final_out(const float* __restrict__ g, float* __restrict__ out) {
  const int gid = blockIdx.x * blockDim.x + threadIdx.x;   // 0..524287
  const int b  = gid >> 16;
  const int hw = gid & 65535;
  const int h  = hw >> 8;
  const int w  = hw & 255;
  const float* gb = g + b * LTOT;
  out[gid] = gb[h] * gb[256 + w] * gb[512 + w] * gb[768 + w];
}

extern "C" void kernel_launch(void* const* d_in, const int* in_sizes, int n_in,
                              void* d_out, int out_size, void* d_ws, size_t ws_size,
                              hipStream_t stream) {
  const float* x        = (const float*)d_in[0];
  const float* w1       = (const float*)d_in[1];
  const float* b1       = (const float*)d_in[2];
  const float* bn_gamma = (const float*)d_in[3];
  const float* bn_beta  = (const float*)d_in[4];
  const float* bn_mean  = (const float*)d_in[5];
  const float* bn_var   = (const float*)d_in[6];
  const float* wh  = (const float*)d_in[7];
  const float* bh  = (const float*)d_in[8];
  const float* ww  = (const float*)d_in[9];
  const float* bw  = (const float*)d_in[10];
  const float* wd1 = (const float*)d_in[11];
  const float* bd1 = (const float*)d_in[12];
  const float* wd2 = (const float*)d_in[13];
  const float* bd2 = (const float*)d_in[14];

  float* ws      = (float*)d_ws;
  float* rowmean = ws;                  // B*C*H   = 524288 f32
  float* colmean = ws + 524288;         // B*C*W   = 524288 f32
  float* y       = ws + 1048576;        // B*MIP*L = 131072 f32
  float* g       = ws + 1179648;        // B*L     =   8192 f32
  float* out     = (float*)d_out;       // B*1*H*W = 524288 f32

  reduce_rc<<<BB * CC, 256, 0, stream>>>(x, rowmean, colmean);
  mlp_wmma<<<64, 256, 0, stream>>>(rowmean, colmean, w1, b1,
                                   bn_gamma, bn_beta, bn_mean, bn_var, y);
  gates_k<<<32, 256, 0, stream>>>(y, wh, bh, ww, bw, wd1, bd1, wd2, bd2, g);
  final_out<<<2048, 256, 0, stream>>>(g, out);
}